// IVPLoss_16544214024787
// MI455X (gfx1250) — compile-verified
//
#include <hip/hip_runtime.h>

// ---------------------------------------------------------------------------
// IVP trajectory MSE loss for MI455X (gfx1250, wave32).
//
// vf_pred / vf_true: (8, 2, 768, 768) f32. 8 Euler steps of bilinear-sampled
// advection per field, then MSE over all (9,8,2,768,768) trajectory coords.
//
// Roofline: both fields (75 MB) are L2-resident (192 MB L2); HBM (23.3 TB/s)
// is irrelevant after first touch. Cost = ~600M data-dependent 4B L2 gathers
// -> gather-latency bound. Strategy: one lane per pixel, trajectory state in
// registers, pred & true gather chains interleaved for ILP, 18K blocks x 8
// wave32s to hide L2 latency. Final reduction is a deterministic f32 chain of
// V_WMMA_F32_16X16X4_F32 (A = ones -> column-sum reducer, 64 values/op).
// ---------------------------------------------------------------------------

typedef __attribute__((ext_vector_type(2))) float v2f;
typedef __attribute__((ext_vector_type(8))) float v8f;

#define Bn 8
#define Hn 768
#define Wn 768
#define HW (Hn * Wn)
#define CHW (2 * HW)
#define NPIX (Bn * HW)          // 4,718,592
#define NSTEPS 8
#define DXC 0.5f
#define BLOCK 256
#define NBLOCKS (NPIX / BLOCK)  // 18,432 (exact; multiple of 64)
#define NCHUNKS (NBLOCKS / 64)  // 288 WMMA reduction iterations

// One Euler step: bilinearly sample both channels of vf at (px,py) (clamped),
// advance the point. 'base' = b*CHW (batch's x-channel plane); y-channel plane
// is at base + HW. 4 shared corner indices, 8 dword gathers.
__device__ __forceinline__ void euler_step(const float* __restrict__ vf,
                                           int base, float& px, float& py) {
    float xc = fminf(fmaxf(px, 0.0f), (float)(Wn - 1));
    float yc = fminf(fmaxf(py, 0.0f), (float)(Hn - 1));
    float xf = floorf(xc);
    float yf = floorf(yc);
    float wx = xc - xf;
    float wy = yc - yf;
    int ix0 = (int)xf;
    int iy0 = (int)yf;
    int ix1 = min(ix0 + 1, Wn - 1);
    int iy1 = min(iy0 + 1, Hn - 1);
    int r0 = base + iy0 * Wn;
    int r1 = base + iy1 * Wn;
    // x-channel corners
    float a00 = vf[r0 + ix0];
    float a01 = vf[r0 + ix1];
    float a10 = vf[r1 + ix0];
    float a11 = vf[r1 + ix1];
    // y-channel corners (next plane, HW elements away)
    float b00 = vf[HW + r0 + ix0];
    float b01 = vf[HW + r0 + ix1];
    float b10 = vf[HW + r1 + ix0];
    float b11 = vf[HW + r1 + ix1];
    float omx = 1.0f - wx;
    float omy = 1.0f - wy;
    float vx = omy * (omx * a00 + wx * a01) + wy * (omx * a10 + wx * a11);
    float vy = omy * (omx * b00 + wx * b01) + wy * (omx * b10 + wx * b11);
    px = fmaf(DXC, vx, px);
    py = fmaf(DXC, vy, py);
}

// Kernel 1: per-pixel trajectory integration for both fields; accumulate
// squared coordinate differences over steps 1..8 (step 0 contributes exactly
// zero: both trajectories start on the identity grid). Deterministic LDS tree
// reduction to one partial per block.
__global__ __launch_bounds__(BLOCK) void ivp_traj_mse(
    const float* __restrict__ vtrue, const float* __restrict__ vpred,
    float* __restrict__ partials) {
    __shared__ float sdata[BLOCK];

    int g = blockIdx.x * BLOCK + threadIdx.x;   // g < NPIX by construction
    int b = g / HW;
    int p = g - b * HW;
    int y = p / Wn;
    int x = p - y * Wn;
    int base = b * CHW;

    float tx = (float)x, ty = (float)y;   // trajectory through vf_true
    float qx = (float)x, qy = (float)y;   // trajectory through vf_pred

    float acc = 0.0f;
#pragma unroll
    for (int s = 0; s < NSTEPS; ++s) {
        // two independent gather chains -> ILP 2 while waiting on L2
        euler_step(vtrue, base, tx, ty);
        euler_step(vpred, base, qx, qy);
        float dx = tx - qx;
        float dy = ty - qy;
        acc = fmaf(dx, dx, acc);
        acc = fmaf(dy, dy, acc);
    }

    sdata[threadIdx.x] = acc;
    __syncthreads();
#pragma unroll
    for (int s = BLOCK / 2; s > 0; s >>= 1) {
        if (threadIdx.x < s) sdata[threadIdx.x] += sdata[threadIdx.x + s];
        __syncthreads();
    }
    if (threadIdx.x == 0) partials[blockIdx.x] = sdata[0];
}

// Kernel 2: single wave32. Reduce NBLOCKS partials 64-at-a-time with
// V_WMMA_F32_16X16X4_F32:  D = ones(16x4) * B(4x16) + C  accumulates the
// column sums of each chunk into C. A = ones makes the reduction invariant to
// any permutation of the 64 chunk values into the B operand, so each lane may
// load two CONSECUTIVE floats as one b64 (coalesced, half the vmem ops), and
// no layout knowledge is needed: sum of all 256 accumulator elements equals
// 16 x grand total (divide by 16 is exact). NBLOCKS is a multiple of 64 ->
// no bounds checks, straight-line loop, EXEC stays all-ones (WMMA-legal).
// Full f32 precision, fixed order -> deterministic.
__global__ __launch_bounds__(32) void reduce_wmma(
    const float* __restrict__ partials, float* __restrict__ out, float scale) {
    int lane = threadIdx.x;  // 0..31

    v8f c = {};              // f32 accumulator (8 VGPRs)
    v2f ones;
    ones.x = 1.0f;
    ones.y = 1.0f;

    const float2* p2 = (const float2*)partials;  // NBLOCKS/2 float2s
#pragma unroll 4
    for (int it = 0; it < NCHUNKS; ++it) {
        float2 v = p2[it * 32 + lane];           // global_load_b64, coalesced
        v2f bv;
        bv.x = v.x;
        bv.y = v.y;
        // args: (neg_a, A, neg_b, B, c_mod, C, reuse_a, reuse_b)
        c = __builtin_amdgcn_wmma_f32_16x16x4_f32(
            false, ones, false, bv, (short)0, c, false, false);
    }

    // Sum of all 256 accumulator elements = 16 * grand total.
    float s = c[0] + c[1] + c[2] + c[3] + c[4] + c[5] + c[6] + c[7];
    __shared__ float red[32];
    red[lane] = s;
    __syncthreads();
    if (lane == 0) {
        double t = 0.0;
        for (int i = 0; i < 32; ++i) t += (double)red[i];
        out[0] = (float)(t * (1.0 / 16.0) * (double)scale);
    }
}

extern "C" void kernel_launch(void* const* d_in, const int* in_sizes, int n_in,
                              void* d_out, int out_size, void* d_ws,
                              size_t ws_size, hipStream_t stream) {
    const float* vf_pred = (const float*)d_in[0];
    const float* vf_true = (const float*)d_in[1];
    float* out = (float*)d_out;
    float* partials = (float*)d_ws;  // NBLOCKS floats = 73,728 B of scratch

    ivp_traj_mse<<<NBLOCKS, BLOCK, 0, stream>>>(vf_true, vf_pred, partials);

    // mean over (NSTEPS+1) * B * 2 * H * W elements (step 0 contributes 0)
    const float scale =
        (float)(1.0 / ((double)(NSTEPS + 1) * Bn * 2.0 * Hn * Wn));
    reduce_wmma<<<1, 32, 0, stream>>>(partials, out, scale);
}